// gru_84550726189862
// MI455X (gfx1250) — compile-verified
//
#include <hip/hip_runtime.h>

// Problem constants (match reference)
#define PADTOK 0
static constexpr int Tt = 50, Bb = 128, Hh = 512, Vv = 30000, DWw = 512;
static constexpr int G3 = 3 * Hh;          // 1536
static constexpr int TB = Tt * Bb;         // 6400
static constexpr int LDA = 516;            // padded LDS row stride (floats)

typedef float v2f __attribute__((ext_vector_type(2)));
typedef float v4f __attribute__((ext_vector_type(4)));
typedef float v8f __attribute__((ext_vector_type(8)));

__device__ __forceinline__ v8f wmma4(v2f a, v2f b, v8f c) {
  // D(16x16,f32) += A(16x4,f32) * B(4x16,f32)
  return __builtin_amdgcn_wmma_f32_16x16x4_f32(false, a, false, b, (short)0, c,
                                               false, false);
}
__device__ __forceinline__ v8f zero8() {
  v8f z = {0.f, 0.f, 0.f, 0.f, 0.f, 0.f, 0.f, 0.f};
  return z;
}
__device__ __forceinline__ v2f lo2(v4f v) {
  return __builtin_shufflevector(v, v, 0, 1);
}
__device__ __forceinline__ v2f hi2(v4f v) {
  return __builtin_shufflevector(v, v, 2, 3);
}
__device__ __forceinline__ v4f ld4(const float* p) { return *(const v4f*)p; }
__device__ __forceinline__ float sigm(float x) {
  return 1.0f / (1.0f + __expf(-x));
}
__device__ __forceinline__ float tanh_fast(float x) {
  return 1.0f - 2.0f / (__expf(2.0f * x) + 1.0f);
}

__global__ void k_zero(float* __restrict__ p, int n) {
  int i = blockIdx.x * blockDim.x + threadIdx.x;
  if (i < n) p[i] = 0.f;
}

// xg[row, n] = sum_k word_emb[review_input[row], k] * W_ih[n, k] + b_ih[n]
// 200 blocks x 32 rows; A tile staged in LDS; 8 waves split the 96 N tiles.
__global__ __launch_bounds__(256) void k_xg(const int* __restrict__ rin,
                                            const float* __restrict__ wemb,
                                            const float* __restrict__ Wih,
                                            const float* __restrict__ bih,
                                            float* __restrict__ xg) {
  extern __shared__ __align__(16) float smem[];   // 32*LDA floats
  __shared__ int toks[32];
  const int m0 = blockIdx.x * 32;
  const int tid = threadIdx.x;
  if (tid < 32) toks[tid] = rin[m0 + tid];
  __syncthreads();
  for (int idx = tid; idx < 32 * 128; idx += 256) {
    const int row = idx >> 7, kq = (idx & 127) << 2;
    *(v4f*)&smem[row * LDA + kq] = ld4(wemb + (size_t)toks[row] * DWw + kq);
  }
  __syncthreads();
  const int lane = tid & 31, wav = tid >> 5;
  const int sel = lane >> 4, lc = lane & 15;
  const unsigned ab0 = (unsigned)(lc * LDA + 4 * sel);
  for (int ti = wav; ti < G3 / 16; ti += 8) {
    unsigned ab = ab0;
    asm volatile("" : "+v"(ab));   // keep per-tile LDS loads in the loop
    const float* a0 = smem + ab;
    const float* a1 = a0 + 16 * LDA;
    const int ncol = ti * 16 + lc;
    const float* bp = Wih + (size_t)ncol * DWw + 4 * sel;
    v4f bA = ld4(bp), bB = ld4(bp + 8);
    v8f c0 = zero8(), c1 = zero8();
#pragma unroll 4
    for (int k = 0; k < DWw; k += 16) {
      v4f b = bA;
      if (k + 16 < DWw) bA = ld4(bp + k + 16);
      v4f x0 = ld4(a0 + k), x1 = ld4(a1 + k);
      c0 = wmma4(lo2(x0), lo2(b), c0);
      c0 = wmma4(hi2(x0), hi2(b), c0);
      c1 = wmma4(lo2(x1), lo2(b), c1);
      c1 = wmma4(hi2(x1), hi2(b), c1);
      b = bB;
      if (k + 24 < DWw) bB = ld4(bp + k + 24);
      x0 = ld4(a0 + k + 8); x1 = ld4(a1 + k + 8);
      c0 = wmma4(lo2(x0), lo2(b), c0);
      c0 = wmma4(hi2(x0), hi2(b), c0);
      c1 = wmma4(lo2(x1), lo2(b), c1);
      c1 = wmma4(hi2(x1), hi2(b), c1);
    }
    const float bb = bih[ncol];
#pragma unroll
    for (int i = 0; i < 8; ++i) {
      const int r0 = m0 + sel * 8 + i;
      const int r1 = r0 + 16;
      xg[(size_t)r0 * G3 + ncol] = c0[i] + bb;
      xg[(size_t)r1 * G3 + ncol] = c1[i] + bb;
    }
  }
}

// One GRU step: reads hprev[128,512], xg_t[128,1536]; writes hcur[128,512].
// 8 blocks x 16 rows; A (h_prev tile) staged in LDS and reused for z*h term.
__global__ __launch_bounds__(256) void k_gru(const float* __restrict__ hprev,
                                             float* __restrict__ hcur,
                                             const float* __restrict__ xg_t,
                                             const float* __restrict__ Whh,
                                             const float* __restrict__ bhh) {
  __shared__ __align__(16) float Al[16 * LDA];
  const int m0 = blockIdx.x * 16;
  const int tid = threadIdx.x;
  for (int idx = tid; idx < 16 * 128; idx += 256) {
    const int row = idx >> 7, kq = (idx & 127) << 2;
    *(v4f*)&Al[row * LDA + kq] = ld4(hprev + (size_t)(m0 + row) * Hh + kq);
  }
  __syncthreads();
  const int lane = tid & 31, wav = tid >> 5;
  const int sel = lane >> 4, lc = lane & 15;
  const unsigned ab0 = (unsigned)(lc * LDA + 4 * sel);
  for (int jt = wav; jt < Hh / 16; jt += 8) {
    unsigned ab = ab0;
    asm volatile("" : "+v"(ab));
    const float* a0 = Al + ab;
    const int j = jt * 16 + lc;  // 0..511 within each gate chunk
    const float* bpr = Whh + (size_t)j * Hh + 4 * sel;
    const float* bpz = bpr + (size_t)Hh * Hh;         // gate z: n = 512 + j
    const float* bpn = bpr + (size_t)2 * Hh * Hh;     // gate n: n = 1024 + j
    v4f rA = ld4(bpr), zA = ld4(bpz), nA = ld4(bpn);
    v4f rB = ld4(bpr + 8), zB = ld4(bpz + 8), nB = ld4(bpn + 8);
    v8f cr = zero8(), cz = zero8(), cn = zero8();
#pragma unroll 2
    for (int k = 0; k < Hh; k += 16) {
      v4f wr = rA, wz = zA, wn = nA;
      if (k + 16 < Hh) {
        rA = ld4(bpr + k + 16); zA = ld4(bpz + k + 16); nA = ld4(bpn + k + 16);
      }
      v4f a = ld4(a0 + k);
      cr = wmma4(lo2(a), lo2(wr), cr);
      cr = wmma4(hi2(a), hi2(wr), cr);
      cz = wmma4(lo2(a), lo2(wz), cz);
      cz = wmma4(hi2(a), hi2(wz), cz);
      cn = wmma4(lo2(a), lo2(wn), cn);
      cn = wmma4(hi2(a), hi2(wn), cn);
      wr = rB; wz = zB; wn = nB;
      if (k + 24 < Hh) {
        rB = ld4(bpr + k + 24); zB = ld4(bpz + k + 24); nB = ld4(bpn + k + 24);
      }
      a = ld4(a0 + k + 8);
      cr = wmma4(lo2(a), lo2(wr), cr);
      cr = wmma4(hi2(a), hi2(wr), cr);
      cz = wmma4(lo2(a), lo2(wz), cz);
      cz = wmma4(hi2(a), hi2(wz), cz);
      cn = wmma4(lo2(a), lo2(wn), cn);
      cn = wmma4(hi2(a), hi2(wn), cn);
    }
    const float br = bhh[j], bz = bhh[Hh + j], bn = bhh[2 * Hh + j];
#pragma unroll
    for (int i = 0; i < 8; ++i) {
      const int lrow = sel * 8 + i;
      const int row = m0 + lrow;
      const size_t xb = (size_t)row * G3;
      const float r = sigm(xg_t[xb + j] + cr[i] + br);
      const float z = sigm(xg_t[xb + Hh + j] + cz[i] + bz);
      const float nn = tanh_fast(xg_t[xb + 2 * Hh + j] + r * (cn[i] + bn));
      const float hp = Al[lrow * LDA + j];   // h_prev from LDS tile
      hcur[(size_t)row * Hh + j] = (1.f - z) * nn + z * hp;
    }
  }
}

// Fused logits GEMM + online log-softmax + target gather + NLL.
// 200 blocks x 32 rows; A tile in dynamic LDS, region reused for reduction.
__global__ __launch_bounds__(256) void k_loss(const float* __restrict__ hseq,
                                              const float* __restrict__ Wout,
                                              const float* __restrict__ bout,
                                              const int* __restrict__ tgt,
                                              float* __restrict__ out_loss) {
  extern __shared__ __align__(16) float smem[];  // 32*LDA floats (>= 2*4096)
  __shared__ float tlog[32];
  __shared__ int ttgt[32];
  const int m0 = blockIdx.x * 32;
  const int tid = threadIdx.x, lane = tid & 31, wav = tid >> 5;
  const int sel = lane >> 4, lc = lane & 15;
  if (tid < 32) {
    ttgt[tid] = tgt[m0 + tid];
    tlog[tid] = 0.f;
  }
  for (int idx = tid; idx < 32 * 128; idx += 256) {
    const int row = idx >> 7, kq = (idx & 127) << 2;
    *(v4f*)&smem[row * LDA + kq] = ld4(hseq + (size_t)(m0 + row) * Hh + kq);
  }
  __syncthreads();
  const unsigned ab0 = (unsigned)(lc * LDA + 4 * sel);
  float mrun[16], srun[16];  // (row-subtile s, vgpr i) -> idx s*8+i
#pragma unroll
  for (int i = 0; i < 16; ++i) {
    mrun[i] = -3.4e38f;
    srun[i] = 0.f;
  }
  for (int ti = wav; ti < Vv / 16; ti += 8) {
    unsigned ab = ab0;
    asm volatile("" : "+v"(ab));
    const float* a0 = smem + ab;
    const float* a1 = a0 + 16 * LDA;
    const int ncol = ti * 16 + lc;
    const float* bp = Wout + (size_t)ncol * Hh + 4 * sel;
    // next tile for this wave (prefetch target); self if this is the last one
    const float* pf = (ti + 8 < Vv / 16) ? (bp + (size_t)128 * Hh) : bp;
    v4f bA = ld4(bp), bB = ld4(bp + 8);
    v8f c0 = zero8(), c1 = zero8();
#pragma unroll 4
    for (int k = 0; k < Hh; k += 16) {
      if ((k & 16) == 0) __builtin_prefetch(pf + k, 0, 0);
      v4f b = bA;
      if (k + 16 < Hh) bA = ld4(bp + k + 16);
      v4f x0 = ld4(a0 + k), x1 = ld4(a1 + k);
      c0 = wmma4(lo2(x0), lo2(b), c0);
      c0 = wmma4(hi2(x0), hi2(b), c0);
      c1 = wmma4(lo2(x1), lo2(b), c1);
      c1 = wmma4(hi2(x1), hi2(b), c1);
      b = bB;
      if (k + 24 < Hh) bB = ld4(bp + k + 24);
      x0 = ld4(a0 + k + 8); x1 = ld4(a1 + k + 8);
      c0 = wmma4(lo2(x0), lo2(b), c0);
      c0 = wmma4(hi2(x0), hi2(b), c0);
      c1 = wmma4(lo2(x1), lo2(b), c1);
      c1 = wmma4(hi2(x1), hi2(b), c1);
    }
    const float bb = bout[ncol];
#pragma unroll
    for (int i = 0; i < 8; ++i) {
      const int lr0 = sel * 8 + i;      // local row, subtile 0
      const int lr1 = 16 + lr0;         // local row, subtile 1
      float x = c0[i] + bb;
      if (x > mrun[i]) { srun[i] *= __expf(mrun[i] - x); mrun[i] = x; }
      srun[i] += __expf(x - mrun[i]);
      if (ncol == ttgt[lr0]) tlog[lr0] = x;
      float y = c1[i] + bb;
      if (y > mrun[8 + i]) { srun[8 + i] *= __expf(mrun[8 + i] - y); mrun[8 + i] = y; }
      srun[8 + i] += __expf(y - mrun[8 + i]);
      if (ncol == ttgt[lr1]) tlog[lr1] = y;
    }
  }
  // A tile is dead: reuse the dynamic LDS region for the reduction arrays.
  __syncthreads();
  float* redm = smem;               // 32*128 floats
  float* reds = smem + 32 * 128;    // 32*128 floats
  const int slot = wav * 16 + lc;   // 0..127 partials per row
#pragma unroll
  for (int i = 0; i < 8; ++i) {
    const int lr0 = sel * 8 + i, lr1 = 16 + lr0;
    redm[lr0 * 128 + slot] = mrun[i];
    reds[lr0 * 128 + slot] = srun[i];
    redm[lr1 * 128 + slot] = mrun[8 + i];
    reds[lr1 * 128 + slot] = srun[8 + i];
  }
  __syncthreads();
  if (tid < 32) {
    float m = -3.4e38f, s = 0.f;
    for (int p = 0; p < 128; ++p) {
      const float pm = redm[tid * 128 + p], ps = reds[tid * 128 + p];
      const float nm = fmaxf(m, pm);
      s = s * __expf(m - nm) + ps * __expf(pm - nm);
      m = nm;
    }
    const float lse = m + __logf(s);
    const int tg = ttgt[tid];
    out_loss[m0 + tid] = (tg == PADTOK) ? 0.f : (lse - tlog[tid]);
  }
}

// obj = sum(out_loss) / max(count(tgt != PAD), 1)
__global__ __launch_bounds__(256) void k_obj(const float* __restrict__ loss,
                                             const int* __restrict__ tgt,
                                             float* __restrict__ obj) {
  __shared__ float ssum[256];
  __shared__ int scnt[256];
  float s = 0.f;
  int c = 0;
  for (int i = threadIdx.x; i < TB; i += 256) {
    s += loss[i];
    c += (tgt[i] != PADTOK) ? 1 : 0;
  }
  ssum[threadIdx.x] = s;
  scnt[threadIdx.x] = c;
  __syncthreads();
  for (int off = 128; off > 0; off >>= 1) {
    if (threadIdx.x < off) {
      ssum[threadIdx.x] += ssum[threadIdx.x + off];
      scnt[threadIdx.x] += scnt[threadIdx.x + off];
    }
    __syncthreads();
  }
  if (threadIdx.x == 0) {
    const int d = scnt[0] < 1 ? 1 : scnt[0];
    obj[0] = ssum[0] / (float)d;
  }
}

extern "C" void kernel_launch(void* const* d_in, const int* in_sizes, int n_in,
                              void* d_out, int out_size, void* d_ws,
                              size_t ws_size, hipStream_t stream) {
  const int* rin = (const int*)d_in[2];    // review_input [T,B]
  const int* rtg = (const int*)d_in[3];    // review_target [T,B]
  const float* wemb = (const float*)d_in[4];
  const float* Wih = (const float*)d_in[5];
  const float* Whh = (const float*)d_in[6];
  const float* bih = (const float*)d_in[7];
  const float* bhh = (const float*)d_in[8];
  const float* Wout = (const float*)d_in[9];
  const float* bout = (const float*)d_in[10];
  float* out = (float*)d_out;              // [6400 out_loss][1 obj]

  float* xg = (float*)d_ws;                         // TB*G3 floats
  float* hseq = xg + (size_t)TB * G3;               // TB*H floats
  float* h0 = hseq + (size_t)TB * Hh;               // B*H floats (zeros)

  const size_t tile_lds = (size_t)32 * LDA * sizeof(float);  // 66,048 B

  k_zero<<<(Bb * Hh + 255) / 256, 256, 0, stream>>>(h0, Bb * Hh);
  k_xg<<<TB / 32, 256, tile_lds, stream>>>(rin, wemb, Wih, bih, xg);
  for (int t = 0; t < Tt; ++t) {
    const float* hp = (t == 0) ? h0 : (hseq + (size_t)(t - 1) * Bb * Hh);
    k_gru<<<Bb / 16, 256, 0, stream>>>(hp, hseq + (size_t)t * Bb * Hh,
                                       xg + (size_t)t * Bb * G3, Whh, bhh);
  }
  k_loss<<<TB / 32, 256, tile_lds, stream>>>(hseq, Wout, bout, rtg, out);
  k_obj<<<1, 256, 0, stream>>>(out, rtg, out + TB);
}